// UI2codeAttention_12326556140282
// MI455X (gfx1250) — compile-verified
//
#include <hip/hip_runtime.h>
#include <math.h>

// Problem constants (match reference)
#define B_    64
#define T_    2048
#define H_    512
#define IN_   512
#define KIH   (IN_ + H_)   // 1024
#define FOURH (4 * H_)     // 2048

typedef __attribute__((ext_vector_type(2))) float v2f;
typedef __attribute__((ext_vector_type(8))) float v8f;

// f32 WMMA: D(16x16) = A(16x4) * B(4x16) + C. 8-arg form:
// (neg_a, A, neg_b, B, c_mod, C, reuse_a, reuse_b)
__device__ __forceinline__ v8f wmma_k4(v2f a, v2f b, v8f c) {
  return __builtin_amdgcn_wmma_f32_16x16x4_f32(false, a, false, b, (short)0, c,
                                               false, false);
}

__device__ __forceinline__ float sigmoidf_(float x) {
  return 1.0f / (1.0f + __expf(-x));
}

// gfx1250 async global->LDS copy, 16 bytes per lane. ASYNCcnt-tracked.
// lds_addr: 32-bit LDS byte address (low 32 bits of generic shared pointer);
// gsrc: 64-bit global address (16B aligned).
__device__ __forceinline__ void async_copy_b128(unsigned lds_addr,
                                                const float* gsrc) {
  asm volatile("global_load_async_to_lds_b128 %0, %1, off"
               :
               : "v"(lds_addr), "v"(gsrc)
               : "memory");
}
__device__ __forceinline__ void async_wait2() {
  asm volatile("s_wait_asynccnt 0x2" ::: "memory");
}
__device__ __forceinline__ void async_wait0() {
  asm volatile("s_wait_asynccnt 0x0" ::: "memory");
}

// -------------------------------------------------------------------------
// Kernel 1: gates[64][2048] = [xt|prev_ctx] @ W_ih^T + prev_h @ W_hh^T
// One wave per 16x16 output tile; 4 waves per block on adjacent N tiles.
// A-lane layout (16x4 f32): lane<16 -> K=0,1 ; lane>=16 -> K=2,3 (float2).
// B-lane layout symmetric: B[K][N] = W[n0+N][k+K] -> same float2 pattern.
// -------------------------------------------------------------------------
__global__ void k_gates(const float* __restrict__ xt,
                        const float* __restrict__ prev_ctx,
                        const float* __restrict__ prev_h,
                        const float* __restrict__ W_ih,
                        const float* __restrict__ W_hh,
                        float* __restrict__ gates) {
  const int lane = threadIdx.x & 31;
  const int wv   = threadIdx.x >> 5;
  const int m0   = blockIdx.y * 16;
  const int n0   = (blockIdx.x * 4 + wv) * 16;
  const int row  = lane & 15;
  const int koff = (lane >> 4) << 1;  // 0 or 2

  const float* ax = xt       + (size_t)(m0 + row) * IN_;
  const float* ac = prev_ctx + (size_t)(m0 + row) * H_;
  const float* ah = prev_h   + (size_t)(m0 + row) * H_;   // prev_h[0] view
  const float* bi = W_ih     + (size_t)(n0 + row) * KIH;
  const float* bh = W_hh     + (size_t)(n0 + row) * H_;

  v8f acc = {};
  for (int k = 0; k < IN_; k += 4) {
    const int kk = k + koff;
    v2f a = {ax[kk], ax[kk + 1]};
    v2f b = {bi[kk], bi[kk + 1]};
    acc = wmma_k4(a, b, acc);
  }
  for (int k = 0; k < H_; k += 4) {
    const int kk = k + koff;
    v2f a = {ac[kk], ac[kk + 1]};
    v2f b = {bi[IN_ + kk], bi[IN_ + kk + 1]};
    acc = wmma_k4(a, b, acc);
  }
  for (int k = 0; k < H_; k += 4) {
    const int kk = k + koff;
    v2f a = {ah[kk], ah[kk + 1]};
    v2f b = {bh[kk], bh[kk + 1]};
    acc = wmma_k4(a, b, acc);
  }

  // C/D layout: VGPR v, lanes 0-15 -> M=v ; lanes 16-31 -> M=v+8 ; N=lane&15
  const int mbase = m0 + ((lane >> 4) << 3);
  const int col   = n0 + (lane & 15);
#pragma unroll
  for (int v = 0; v < 8; ++v)
    gates[(size_t)(mbase + v) * FOURH + col] = acc[v];
}

// -------------------------------------------------------------------------
// Kernel 2: LSTM pointwise -> h, c
// -------------------------------------------------------------------------
__global__ void k_lstm(const float* __restrict__ gates,
                       const float* __restrict__ prev_c,
                       float* __restrict__ h_out,
                       float* __restrict__ c_out) {
  const int idx = blockIdx.x * blockDim.x + threadIdx.x;  // 0..32767
  const int b = idx >> 9;
  const int j = idx & (H_ - 1);
  const float* g = gates + (size_t)b * FOURH;
  const float gi = g[j];
  const float gf = g[H_ + j];
  const float gg = g[2 * H_ + j];
  const float go = g[3 * H_ + j];
  const float c = sigmoidf_(gf) * prev_c[idx] + sigmoidf_(gi) * tanhf(gg);
  const float h = sigmoidf_(go) * tanhf(c);
  h_out[idx] = h;
  c_out[idx] = c;
}

// -------------------------------------------------------------------------
// Kernel 3: mapped_h = h @ W_hidden^T  (M=64, N=512, K=512)
// -------------------------------------------------------------------------
__global__ void k_maph(const float* __restrict__ h,
                       const float* __restrict__ W_hidden,
                       float* __restrict__ mapped) {
  const int lane = threadIdx.x & 31;
  const int wv   = threadIdx.x >> 5;
  const int m0   = blockIdx.y * 16;
  const int n0   = (blockIdx.x * 4 + wv) * 16;
  const int row  = lane & 15;
  const int koff = (lane >> 4) << 1;

  const float* ar = h        + (size_t)(m0 + row) * H_;
  const float* br = W_hidden + (size_t)(n0 + row) * H_;

  v8f acc = {};
  for (int k = 0; k < H_; k += 4) {
    const int kk = k + koff;
    v2f a = {ar[kk], ar[kk + 1]};
    v2f b = {br[kk], br[kk + 1]};
    acc = wmma_k4(a, b, acc);
  }
  const int mbase = m0 + ((lane >> 4) << 3);
  const int col   = n0 + (lane & 15);
#pragma unroll
  for (int v = 0; v < 8; ++v)
    mapped[(size_t)(mbase + v) * H_ + col] = acc[v];
}

// -------------------------------------------------------------------------
// Kernel 4: fused attention (flash-style, single context pass) with
// double-buffered GLOBAL_LOAD_ASYNC_TO_LDS_B128 staging of context tiles.
//   attn[b,t] = context[b,t,:] . mapped_h[b,:]
//   ctx_comb  = softmax_t(attn) @ context[b]
// One block per batch; 512 threads = 16 waves. Tile = 8 rows x 512 cols
// (16 KB). Row r's logit computed by waves 2r/2r+1 (256-col halves);
// every thread owns one H-column of the running weighted sum.
// -------------------------------------------------------------------------
#define RT 8               // context rows per tile
#define NT (T_ / RT)       // 256 tiles

__global__ void k_attn(const float* __restrict__ context,
                       const float* __restrict__ mapped,
                       float* __restrict__ ctx_comb) {
  __shared__ __align__(16) float tile[2][RT * H_];  // 2 x 16 KB
  __shared__ float mh[H_];
  __shared__ float red2[16];

  const int b    = blockIdx.x;
  const int tid  = threadIdx.x;   // 0..511
  const int lane = tid & 31;
  const int wv   = tid >> 5;      // 0..15
  const float* ctxb = context + (size_t)b * T_ * H_;

  mh[tid] = mapped[(size_t)b * H_ + tid];

  // LDS byte addresses of the two staging buffers for this thread's slices.
  const unsigned lds0 = (unsigned)(size_t)(&tile[0][0]);
  const unsigned lds1 = (unsigned)(size_t)(&tile[1][0]);

  // Issue one tile: RT*H_ floats = 16 KB = 1024 x b128; 512 threads -> 2 each.
  // Thread tid moves float4 slots {tid, tid+512}.
  auto issue_tile = [&](int buf, int t0) {
    const float* src = ctxb + (size_t)t0 * H_;
    const unsigned base = buf ? lds1 : lds0;
#pragma unroll
    for (int q = 0; q < 2; ++q) {
      const unsigned slot = (unsigned)tid + (unsigned)q * 512u;  // float4 slot
      async_copy_b128(base + slot * 16u, src + (size_t)slot * 4u);
    }
  };

  issue_tile(0, 0);

  float acc = 0.0f;
  float m = -INFINITY;
  float s = 0.0f;

  for (int t = 0; t < NT; ++t) {
    if (t + 1 < NT) {
      issue_tile((t + 1) & 1, (t + 1) * RT);
      async_wait2();   // in-order completion => current tile's 2 ops done
    } else {
      async_wait0();
    }
    __syncthreads();   // all waves' async writes for tile t are visible

    const float* tl = tile[t & 1];

    // logits: row r = wv>>1, column half ch = wv&1
    const int r  = wv >> 1;
    const int ch = wv & 1;
    const float* rowp = tl + r * H_ + ch * 256;
    const float* mhp  = mh + ch * 256;
    float p = 0.0f;
#pragma unroll
    for (int c = lane; c < 256; c += 32) p = fmaf(rowp[c], mhp[c], p);
#pragma unroll
    for (int off = 16; off > 0; off >>= 1) p += __shfl_xor(p, off);
    if (lane == 0) red2[wv] = p;
    __syncthreads();

    // online softmax update (identical on all threads)
    float lg[RT];
    float mnew = m;
#pragma unroll
    for (int rr = 0; rr < RT; ++rr) {
      lg[rr] = red2[2 * rr] + red2[2 * rr + 1];
      mnew = fmaxf(mnew, lg[rr]);
    }
    const float rs = __expf(m - mnew);
    acc *= rs;
    s *= rs;
#pragma unroll
    for (int rr = 0; rr < RT; ++rr) {
      const float w = __expf(lg[rr] - mnew);
      s += w;
      acc = fmaf(w, tl[rr * H_ + tid], acc);
    }
    m = mnew;
    __syncthreads();  // finish reads of tile (t&1) before it is re-filled
  }

  ctx_comb[(size_t)b * H_ + tid] = acc / s;
}

// -------------------------------------------------------------------------
// Kernel 5: out = tanh([ctx_comb|h] @ W_out^T)  (M=64, N=512, K=1024)
// Written to both output slots (reference returns context_output twice).
// -------------------------------------------------------------------------
__global__ void k_out(const float* __restrict__ ctx_comb,
                      const float* __restrict__ h,
                      const float* __restrict__ W_out,
                      float* __restrict__ out0,
                      float* __restrict__ out1) {
  const int lane = threadIdx.x & 31;
  const int wv   = threadIdx.x >> 5;
  const int m0   = blockIdx.y * 16;
  const int n0   = (blockIdx.x * 4 + wv) * 16;
  const int row  = lane & 15;
  const int koff = (lane >> 4) << 1;

  const float* a0 = ctx_comb + (size_t)(m0 + row) * H_;  // k in [0,512)
  const float* a1 = h        + (size_t)(m0 + row) * H_;  // k in [512,1024)
  const float* br = W_out    + (size_t)(n0 + row) * (2 * H_);

  v8f acc = {};
  for (int k = 0; k < H_; k += 4) {
    const int kk = k + koff;
    v2f a = {a0[kk], a0[kk + 1]};
    v2f b = {br[kk], br[kk + 1]};
    acc = wmma_k4(a, b, acc);
  }
  for (int k = 0; k < H_; k += 4) {
    const int kk = k + koff;
    v2f a = {a1[kk], a1[kk + 1]};
    v2f b = {br[H_ + kk], br[H_ + kk + 1]};
    acc = wmma_k4(a, b, acc);
  }

  const int mbase = m0 + ((lane >> 4) << 3);
  const int col   = n0 + (lane & 15);
#pragma unroll
  for (int v = 0; v < 8; ++v) {
    const float val = tanhf(acc[v]);
    const size_t o = (size_t)(mbase + v) * H_ + col;
    out0[o] = val;
    out1[o] = val;
  }
}

// -------------------------------------------------------------------------
extern "C" void kernel_launch(void* const* d_in, const int* in_sizes, int n_in,
                              void* d_out, int out_size, void* d_ws,
                              size_t ws_size, hipStream_t stream) {
  const float* xt       = (const float*)d_in[0];
  const float* context  = (const float*)d_in[1];
  const float* prev_h   = (const float*)d_in[2];  // [1,B,H] == [B,H]
  const float* prev_c   = (const float*)d_in[3];
  const float* prev_ctx = (const float*)d_in[4];
  const float* W_ih     = (const float*)d_in[5];
  const float* W_hh     = (const float*)d_in[6];
  const float* W_hidden = (const float*)d_in[7];
  const float* W_out    = (const float*)d_in[8];

  float* out = (float*)d_out;
  float* out_ctx0 = out;                 // context_output
  float* out_h    = out + B_ * H_;       // h[None]
  float* out_c    = out + 2 * B_ * H_;   // c[None]
  float* out_ctx1 = out + 3 * B_ * H_;   // context_output (again)

  float* ws       = (float*)d_ws;
  float* gates    = ws;                          // 64*2048 f32 = 512 KB
  float* mapped   = gates + (size_t)B_ * FOURH;  // 64*512
  float* ctxcomb  = mapped + (size_t)B_ * H_;    // 64*512

  k_gates<<<dim3(32, B_ / 16), 128, 0, stream>>>(xt, prev_ctx, prev_h, W_ih,
                                                 W_hh, gates);
  k_lstm<<<dim3(B_), 512, 0, stream>>>(gates, prev_c, out_h, out_c);
  k_maph<<<dim3(8, B_ / 16), 128, 0, stream>>>(out_h, W_hidden, mapped);
  k_attn<<<dim3(B_), 512, 0, stream>>>(context, mapped, ctxcomb);
  k_out<<<dim3(8, B_ / 16), 128, 0, stream>>>(ctxcomb, out_h, W_out, out_ctx0,
                                              out_ctx1);
}